// PaiNN_962072674900
// MI455X (gfx1250) — compile-verified
//
#include <hip/hip_runtime.h>
#include <math.h>

#define NE_ 512     // electrons
#define NN_ 64      // nuclei
#define DD_ 128
#define ACCW 132    // per-receiver accumulator width: 32 basis + 96 basis*dir + 3 dir + 1 count

typedef __attribute__((ext_vector_type(2))) float v2f;
typedef __attribute__((ext_vector_type(8))) float v8f;

__device__ __forceinline__ v8f wmma4(v2f a, v2f b, v8f c) {
  // D = A(16x4 f32) x B(4x16 f32) + C(16x16 f32)
  return __builtin_amdgcn_wmma_f32_16x16x4_f32(false, a, false, b, (short)0, c, false, false);
}

__device__ __forceinline__ void atomAddF(float* p, float v) {
  __hip_atomic_fetch_add(p, v, __ATOMIC_RELAXED, __HIP_MEMORY_SCOPE_AGENT);
}
__device__ __forceinline__ void atomAddLds(float* p, float v) {
  __hip_atomic_fetch_add(p, v, __ATOMIC_RELAXED, __HIP_MEMORY_SCOPE_WORKGROUP);
}

// DistanceBasis(DF=32, cutoff=10, 'nocusp', offset=True)
__device__ __forceinline__ float basis_k(int k, float d, float env) {
  const float delta = 1.0f / 64.0f;
  float q  = delta + (float)k * ((1.0f - 2.0f * delta) / 31.0f);
  float mu = 10.0f * q * q;
  float sg = (1.0f + 10.0f * q) * (1.0f / 7.0f);
  float t  = (d - mu) / sg;
  return env * __expf(-t * t);
}

// ---------------- generic WMMA f32 GEMM (branchless loads) ----------------
// C[m*ldc + n*sc] = act( sum_k A[m*lda + k*SA] * B[k*ldb + n] + bias[n] )
// Out-of-range rows/cols are CLAMPED for loads (they only feed unstored C
// elements); only the K tail needs value masking.
template <int SA>
__global__ void k_gemm(const float* __restrict__ A, const float* __restrict__ B,
                       const float* __restrict__ bias, float* __restrict__ C,
                       int M, int N, int K, int lda, int ldb, int ldc, int sc, int act)
{
  const int lane = threadIdx.x;
  const int half = lane >> 4;
  const int l16  = lane & 15;
  const int tN = blockIdx.x, tM = blockIdx.y;
  const int arow = min(tM * 16 + l16, M - 1);   // clamped: safe to load, unstored if padded
  const int bcol = min(tN * 16 + l16, N - 1);
  const float* Ap = A + (size_t)arow * lda + (size_t)(2 * half) * SA;
  const float* Bp = B + (size_t)(2 * half) * ldb + bcol;
  v8f acc = {0.f,0.f,0.f,0.f,0.f,0.f,0.f,0.f};
  int k = 0;
#pragma unroll 2
  for (; k + 4 <= K; k += 4) {
    v2f a, b;
    a.x = Ap[0];  a.y = Ap[SA];
    b.x = Bp[0];  b.y = Bp[ldb];
    Ap += 4 * SA;
    Bp += 4 * (size_t)ldb;
    acc = wmma4(a, b, acc);
  }
  if (k < K) {  // masked tail (K % 4 != 0), branchless via safe-address select
    const int k0 = k + 2 * half;
    const bool m0 = (k0 < K), m1 = (k0 + 1 < K);
    const float* ap0 = m0 ? Ap        : A;
    const float* ap1 = m1 ? (Ap + SA) : A;
    const float* bp0 = m0 ? Bp        : B;
    const float* bp1 = m1 ? (Bp + ldb) : B;
    float ax = *ap0, ay = *ap1, bx = *bp0, by = *bp1;
    v2f a, b;
    a.x = m0 ? ax : 0.f;  a.y = m1 ? ay : 0.f;
    b.x = m0 ? bx : 0.f;  b.y = m1 ? by : 0.f;
    acc = wmma4(a, b, acc);
  }
  const int ccol = tN * 16 + l16;
  if (ccol >= N) return;
  const float bi = bias ? bias[ccol] : 0.f;
#pragma unroll
  for (int v = 0; v < 8; ++v) {
    int crow = tM * 16 + v + 8 * half;
    if (crow >= M) continue;
    float x = acc[v] + bi;
    if (act == 1) x = x / (1.f + __expf(-x));   // silu
    C[(size_t)crow * ldc + (size_t)ccol * sc] = x;
  }
}

// --------- same/anti: LDS-privatized per-edge basis accumulation ----------
// Whole 512x132 receiver accumulator (270KB) lives in one WGP's 320KB LDS.
__global__ void k_edge_const_lds(const float* __restrict__ rs, const int* __restrict__ snd,
                                 const int* __restrict__ rcv, int E, int epw,
                                 float* __restrict__ acc)
{
  extern __shared__ float sacc[];            // NE_*ACCW floats
  const int tid = threadIdx.x;               // 256 threads
  for (int i = tid; i < NE_ * ACCW; i += 256) sacc[i] = 0.f;
  __syncthreads();
  const int e0 = blockIdx.x * epw;
  const int e1 = min(e0 + epw, E);
  for (int e = e0 + tid; e < e1; e += 256) {
    int s = snd[e], r = rcv[e];
    float dx = rs[r*3+0] - rs[s*3+0];
    float dy = rs[r*3+1] - rs[s*3+1];
    float dz = rs[r*3+2] - rs[s*3+2];
    float d2 = dx*dx + dy*dy + dz*dz;
    float d  = sqrtf(d2);
    const float eps = 1.1920929e-7f;
    float inv = 1.f / ((d > eps) ? d : eps);
    float ux = dx*inv, uy = dy*inv, uz = dz*inv;
    float env = d2 * __expf(-d);
    float* a = sacc + r * ACCW;
    for (int k = 0; k < 32; ++k) {
      float bs = basis_k(k, d, env);
      atomAddLds(a + k,            bs);
      atomAddLds(a + 32 + k*3 + 0, bs * ux);
      atomAddLds(a + 32 + k*3 + 1, bs * uy);
      atomAddLds(a + 32 + k*3 + 2, bs * uz);
    }
    atomAddLds(a + 128, ux); atomAddLds(a + 129, uy); atomAddLds(a + 130, uz);
    atomAddLds(a + 131, 1.f);
  }
  __syncthreads();
  for (int i = tid; i < NE_ * ACCW; i += 256) {
    float v = sacc[i];
    if (v != 0.f) atomAddF(acc + i, v);
  }
}

// combine WMMA-produced raw sums with bias*count terms and hx scaling
// z_s[r,d]   = hE[d]    *(ZSr[r,d]   + cnt[r]    * b[d])
// z_v[r,d,i] = hE[256+d]*(ZVr[r,d,i] + sumdir[r,i]*b[256+d])
__global__ void k_finalize2(const float* __restrict__ acc, const float* __restrict__ ZSr,
                            const float* __restrict__ ZVr, const float* __restrict__ b,
                            const float* __restrict__ hE,
                            float* __restrict__ z_s, float* __restrict__ z_v)
{
  int idx = blockIdx.x * blockDim.x + threadIdx.x;
  if (idx >= NE_ * DD_) return;
  int r = idx >> 7, dd = idx & 127;
  const float* a = acc + (size_t)r * ACCW;
  z_s[idx] = hE[dd] * (ZSr[idx] + a[131] * b[dd]);
  float hv = hE[256 + dd], bv = b[256 + dd];
  const float* zr = ZVr + (size_t)idx * 3;
  float* zp = z_v + (size_t)idx * 3;
  zp[0] = hv * (zr[0] + a[128] * bv);
  zp[1] = hv * (zr[1] + a[129] * bv);
  zp[2] = hv * (zr[2] + a[130] * bv);
}

// ---------------- ne: basis precompute ----------------
__global__ void k_ne_basis(const float* __restrict__ rs, const float* __restrict__ coords,
                           const int* __restrict__ snd, const int* __restrict__ rcv,
                           float* __restrict__ basis, float* __restrict__ dir, int E)
{
  int e = blockIdx.x * blockDim.x + threadIdx.x;
  if (e >= E) return;
  int s = snd[e], r = rcv[e];
  float dx = rs[r*3+0] - coords[s*3+0];
  float dy = rs[r*3+1] - coords[s*3+1];
  float dz = rs[r*3+2] - coords[s*3+2];
  float d2 = dx*dx + dy*dy + dz*dz;
  float d  = sqrtf(d2);
  const float eps = 1.1920929e-7f;
  float inv = 1.f / ((d > eps) ? d : eps);
  dir[e*3+0] = dx*inv; dir[e*3+1] = dy*inv; dir[e*3+2] = dz*inv;
  float env = d2 * __expf(-d);
  float* bpp = basis + (size_t)e * 32;
  for (int k = 0; k < 32; ++k) bpp[k] = basis_k(k, d, env);
}

// pack W[:,0:128] | W[:,256:384] -> Wp[32,256]; same for bias
__global__ void k_pack_wp(const float* __restrict__ W, const float* __restrict__ b,
                          float* __restrict__ Wp, float* __restrict__ bp)
{
  int idx = blockIdx.x * blockDim.x + threadIdx.x;
  if (idx < 32 * 256) {
    int k = idx >> 8, c = idx & 255;
    Wp[idx] = W[k*384 + (c < 128 ? c : c + 128)];
  }
  if (idx < 256) bp[idx] = b[idx < 128 ? idx : idx + 128];
}

// ne: fused WMMA (E x 32 @ 32 x 256) + per-element hx multiply + atomic scatter
__global__ void k_ne_scatter(const float* __restrict__ basis, const float* __restrict__ dir,
                             const int* __restrict__ snd, const int* __restrict__ rcv,
                             const float* __restrict__ Wp, const float* __restrict__ bp,
                             const float* __restrict__ hN,
                             float* __restrict__ z_s, float* __restrict__ z_v, int E)
{
  const int lane = threadIdx.x;
  const int half = lane >> 4, l16 = lane & 15;
  const int tN = blockIdx.x, tM = blockIdx.y;
  const int arow = min(tM * 16 + l16, E - 1);      // clamped, branchless loads
  const int bcol = tN * 16 + l16;
  const float* Ar = basis + (size_t)arow * 32 + 2 * half;
  const float* Bp = Wp + (size_t)(2 * half) * 256 + bcol;
  v8f acc = {0.f,0.f,0.f,0.f,0.f,0.f,0.f,0.f};
#pragma unroll
  for (int k = 0; k < 32; k += 4) {
    v2f a, b;
    a.x = Ar[0];   a.y = Ar[1];
    b.x = Bp[0];   b.y = Bp[256];
    Ar += 4;
    Bp += 4 * 256;
    acc = wmma4(a, b, acc);
  }
  const int col = bcol;
  const float bia = bp[col];
#pragma unroll
  for (int v = 0; v < 8; ++v) {
    int e = tM * 16 + v + 8 * half;
    if (e >= E) continue;
    int r = rcv[e], s = snd[e];
    float val = acc[v] + bia;
    if (col < 128) {
      atomAddF(&z_s[(size_t)r * 128 + col], val * hN[(size_t)s * 384 + col]);
    } else {
      int dd = col - 128;
      float pv = val * hN[(size_t)s * 384 + 256 + dd];
      const float* de = dir + (size_t)e * 3;
      float* zp = z_v + ((size_t)r * 128 + dd) * 3;
      atomAddF(zp + 0, pv * de[0]);
      atomAddF(zp + 1, pv * de[1]);
      atomAddF(zp + 2, pv * de[2]);
    }
  }
}

// ---------------- small elementwise kernels ----------------
__global__ void k_pack_xh(const float* __restrict__ X, const float* __restrict__ Y,
                          float* __restrict__ Xh)
{
  int idx = blockIdx.x * blockDim.x + threadIdx.x;
  if (idx >= 65 * 128) return;
  int row = idx / 128, d = idx % 128;
  Xh[idx] = (row == 0) ? X[d] : Y[(row - 1) * 128 + d];
}

__global__ void k_build_xg(const float* __restrict__ z_s, const float* __restrict__ Vv,
                           float* __restrict__ Xg)
{
  int idx = blockIdx.x * blockDim.x + threadIdx.x;
  if (idx >= NE_ * DD_) return;
  int r = idx >> 7, dd = idx & 127;
  Xg[(size_t)r * 256 + dd] = z_s[idx];
  const float* v = Vv + (size_t)idx * 3;
  Xg[(size_t)r * 256 + 128 + dd] = sqrtf(v[0]*v[0] + v[1]*v[1] + v[2]*v[2] + 1e-12f);
}

__global__ void k_combine(const float* __restrict__ G, const float* __restrict__ Uv,
                          const float* __restrict__ Vv, float* __restrict__ upd_s,
                          float* __restrict__ upd_v)
{
  int idx = blockIdx.x * blockDim.x + threadIdx.x;
  if (idx >= NE_ * DD_) return;
  int r = idx >> 7, dd = idx & 127;
  const float* g = G + (size_t)r * 384;
  float a_ss = g[dd], a_vv = g[128 + dd], a_sv = g[256 + dd];
  const float* u  = Uv + (size_t)idx * 3;
  const float* vv = Vv + (size_t)idx * 3;
  float dot = u[0]*vv[0] + u[1]*vv[1] + u[2]*vv[2];
  upd_s[idx] += a_ss + a_sv * dot;
  float* uo = upd_v + (size_t)idx * 3;
  uo[0] += u[0]*a_vv; uo[1] += u[1]*a_vv; uo[2] += u[2]*a_vv;
}

__global__ void k_write_out(const float* __restrict__ X, const float* __restrict__ upd_s,
                            const float* __restrict__ upd_v, float* __restrict__ out)
{
  int idx = blockIdx.x * blockDim.x + threadIdx.x;
  if (idx >= NE_ * DD_) return;
  int r = idx >> 7, dd = idx & 127;
  out[(size_t)r * 512 + dd] = X[dd] + upd_s[idx];
  const float* uv = upd_v + (size_t)idx * 3;
  float* o = out + (size_t)r * 512 + 128 + dd * 3;
  o[0] = uv[0]; o[1] = uv[1]; o[2] = uv[2];
}

// ---------------- host launcher ----------------
extern "C" void kernel_launch(void* const* d_in, const int* in_sizes, int n_in,
                              void* d_out, int out_size, void* d_ws, size_t ws_size,
                              hipStream_t stream)
{
  (void)n_in; (void)out_size;
  // insertion-order flatten of setup_inputs():
  // 0 rs, 1 coords, 2 X, 3 Y, 4 hW1, 5 hb1, 6 hW2, 7 hb2,
  // per label (same,anti,ne,en,nn) x {wW,wb,gW1,gb1,gW2,gb2,V,U} -> bases 8,16,24,32,40
  // then senders/receivers pairs at 48..57
  const float* rs     = (const float*)d_in[0];
  const float* coords = (const float*)d_in[1];
  const float* Xemb   = (const float*)d_in[2];
  const float* Yemb   = (const float*)d_in[3];
  const float* hW1 = (const float*)d_in[4];
  const float* hb1 = (const float*)d_in[5];
  const float* hW2 = (const float*)d_in[6];
  const float* hb2 = (const float*)d_in[7];
  auto P = [&](int base, int o) { return (const float*)d_in[base + o]; };
  const int* snd_same = (const int*)d_in[48]; const int* rcv_same = (const int*)d_in[49];
  const int* snd_anti = (const int*)d_in[50]; const int* rcv_anti = (const int*)d_in[51];
  const int* snd_ne   = (const int*)d_in[52]; const int* rcv_ne   = (const int*)d_in[53];
  const int E_same = in_sizes[48], E_anti = in_sizes[50], E_ne = in_sizes[52];
  float* out = (float*)d_out;

  // workspace carve (floats)
  float* w = (float*)d_ws;
  size_t off = 0;
  auto carve = [&](size_t n) { float* p = w + off; off += (n + 3) & ~(size_t)3; return p; };
  float* acc    = carve(NE_ * ACCW);
  float* z_s    = carve(512 * 128);
  float* z_v    = carve(512 * 384);
  float* ZSr    = carve(512 * 128);
  float* ZVr    = carve(512 * 384);
  float* Vv     = carve(512 * 384);
  float* Uv     = carve(512 * 384);
  float* Xg     = carve(512 * 256);
  float* Hb     = carve(512 * 314);
  float* Gb     = carve(512 * 384);
  float* upd_s  = carve(512 * 128);
  float* upd_v  = carve(512 * 384);
  float* Xh     = carve(80 * 128);
  float* Hh     = carve(80 * 224);
  float* hsrc   = carve(80 * 384);
  float* basisn = carve((size_t)E_ne * 32);
  float* dirn   = carve((size_t)E_ne * 3);
  float* Wp     = carve(32 * 256);
  float* bpk    = carve(256);
  if (ws_size < off * sizeof(float)) return;

  auto gemm = [&](const float* A, const float* B, const float* bias, float* C,
                  int M, int N, int K, int lda, int sa, int ldb, int ldc, int sc, int act) {
    dim3 grid((N + 15) / 16, (M + 15) / 16);
    if (sa == 1)
      k_gemm<1><<<grid, 32, 0, stream>>>(A, B, bias, C, M, N, K, lda, ldb, ldc, sc, act);
    else
      k_gemm<3><<<grid, 32, 0, stream>>>(A, B, bias, C, M, N, K, lda, ldb, ldc, sc, act);
  };

  hipMemsetAsync(upd_s, 0, 512 * 128 * sizeof(float), stream);
  hipMemsetAsync(upd_v, 0, 512 * 384 * sizeof(float), stream);

  // h MLP over the 65 distinct sender rows (1 electron + 64 nuclei)
  k_pack_xh<<<(65 * 128 + 255) / 256, 256, 0, stream>>>(Xemb, Yemb, Xh);
  gemm(Xh, hW1, hb1, Hh,   65, 222, 128, 128, 1, 222, 222, 1, 1);
  gemm(Hh, hW2, hb2, hsrc, 65, 384, 222, 222, 1, 384, 384, 1, 0);
  const float* hE = hsrc;        // electron h(x) row
  const float* hN = hsrc + 384;  // 64 nuclear rows

  auto tail = [&](int base) {
    const float* Vm = P(base, 6);
    const float* Um = P(base, 7);
    for (int i = 0; i < 3; ++i) {
      gemm(z_v + i, Vm, nullptr, Vv + i, 512, 128, 128, 384, 3, 128, 384, 3, 0);
      gemm(z_v + i, Um, nullptr, Uv + i, 512, 128, 128, 384, 3, 128, 384, 3, 0);
    }
    k_build_xg<<<(512 * 128 + 255) / 256, 256, 0, stream>>>(z_s, Vv, Xg);
    gemm(Xg, P(base, 2), P(base, 3), Hb, 512, 314, 256, 256, 1, 314, 314, 1, 1);
    gemm(Hb, P(base, 4), P(base, 5), Gb, 512, 384, 314, 314, 1, 384, 384, 1, 0);
    k_combine<<<(512 * 128 + 255) / 256, 256, 0, stream>>>(Gb, Uv, Vv, upd_s, upd_v);
  };

  // same + anti (constant hx -> factorized edge accumulation via LDS privatization)
  const int bases[2] = {8, 16};
  const int* ss[2] = {snd_same, snd_anti};
  const int* rr[2] = {rcv_same, rcv_anti};
  const int EE[2] = {E_same, E_anti};
  const int EPW = 2048;                                // edges per workgroup
  const size_t ldsBytes = (size_t)NE_ * ACCW * sizeof(float);   // 270336 B < 320KB
  for (int li = 0; li < 2; ++li) {
    hipMemsetAsync(acc, 0, NE_ * ACCW * sizeof(float), stream);
    int nwg = (EE[li] + EPW - 1) / EPW;
    k_edge_const_lds<<<nwg, 256, ldsBytes, stream>>>(rs, ss[li], rr[li], EE[li], EPW, acc);
    const float* Wl = P(bases[li], 0);
    const float* bl = P(bases[li], 1);
    // raw sums via WMMA: ZSr = acc_b @ W[:,0:128]; ZVr_i = acc_bd_i @ W[:,256:384]
    gemm(acc, Wl, nullptr, ZSr, 512, 128, 32, ACCW, 1, 384, 128, 1, 0);
    for (int i = 0; i < 3; ++i)
      gemm(acc + 32 + i, Wl + 256, nullptr, ZVr + i, 512, 128, 32, ACCW, 3, 384, 384, 3, 0);
    k_finalize2<<<(512 * 128 + 255) / 256, 256, 0, stream>>>(acc, ZSr, ZVr, bl, hE, z_s, z_v);
    tail(bases[li]);
  }

  // ne (nucleus -> electron): fused WMMA GEMM + scatter
  hipMemsetAsync(z_s, 0, 512 * 128 * sizeof(float), stream);
  hipMemsetAsync(z_v, 0, 512 * 384 * sizeof(float), stream);
  k_ne_basis<<<(E_ne + 255) / 256, 256, 0, stream>>>(rs, coords, snd_ne, rcv_ne, basisn, dirn, E_ne);
  k_pack_wp<<<(32 * 256 + 255) / 256, 256, 0, stream>>>(P(24, 0), P(24, 1), Wp, bpk);
  {
    dim3 g(16, (E_ne + 15) / 16);
    k_ne_scatter<<<g, 32, 0, stream>>>(basisn, dirn, snd_ne, rcv_ne, Wp, bpk, hN, z_s, z_v, E_ne);
  }
  tail(24);

  // en / nn labels only update nuclear state, which is not part of the output -> skipped.
  k_write_out<<<(512 * 128 + 255) / 256, 256, 0, stream>>>(Xemb, upd_s, upd_v, out);
}